// InteractionBlock_50861002719957
// MI455X (gfx1250) — compile-verified
//
#include <hip/hip_runtime.h>
#include <math.h>

#define NNODES 10000
#define NEDGES 131072

typedef __attribute__((ext_vector_type(2))) float v2f;
typedef __attribute__((ext_vector_type(8))) float v8f;

__device__ inline v8f wmma4(v2f a, v2f b, v8f c) {
    // D = A(16x4,f32) * B(4x16,f32) + C(16x16,f32)
    return __builtin_amdgcn_wmma_f32_16x16x4_f32(false, a, false, b, (short)0, c,
                                                 false, false);
}

// ---------------------------------------------------------------------------
// Wigner-3j constants (exact reproduction of reference _su2_cg/_q/_w3j, fp64)
// ---------------------------------------------------------------------------
struct cd { double re, im; };
__device__ inline cd cmul(cd a, cd b) {
    cd r; r.re = a.re * b.re - a.im * b.im; r.im = a.re * b.im + a.im * b.re; return r;
}
__device__ inline double dfact(int n) { double r = 1.0; for (int i = 2; i <= n; ++i) r *= (double)i; return r; }

__device__ void qmat(int l, cd Q[7][7]) {
    for (int a = 0; a < 7; ++a) for (int b = 0; b < 7; ++b) { Q[a][b].re = 0.0; Q[a][b].im = 0.0; }
    const double is2 = 0.70710678118654752440;
    for (int m = -l; m < 0; ++m) {
        Q[l + m][l - m].re = is2;     // q[l+m, l+|m|] = 1/sqrt2
        Q[l + m][l + m].im = -is2;    // q[l+m, l-|m|] = -i/sqrt2
    }
    Q[l][l].re = 1.0;
    for (int m = 1; m <= l; ++m) {
        double sgn = (m & 1) ? -1.0 : 1.0;
        Q[l + m][l + m].re = sgn * is2;
        Q[l + m][l - m].im = sgn * is2;
    }
    cd f;
    switch (l & 3) {                  // (-i)^l
        case 0:  f.re = 1.0;  f.im = 0.0;  break;
        case 1:  f.re = 0.0;  f.im = -1.0; break;
        case 2:  f.re = -1.0; f.im = 0.0;  break;
        default: f.re = 0.0;  f.im = 1.0;  break;
    }
    for (int a = 0; a < 7; ++a) for (int b = 0; b < 7; ++b) Q[a][b] = cmul(Q[a][b], f);
}

__global__ void w3j_kernel(float* __restrict__ w3j_out) {
    const int L1T[10] = {0,1,0,1,1,0,1,1,0,1};
    const int L2T[10] = {0,1,1,0,2,2,1,3,3,2};
    const int L3T[10] = {0,0,1,1,1,2,2,2,3,3};
    const int W3OFF[10] = {0,1,10,19,28,73,98,143,248,297};
    int k = threadIdx.x;
    if (k >= 10) return;
    int j1 = L1T[k], j2 = L2T[k], j3 = L3T[k];
    int n1 = 2 * j1 + 1, n2 = 2 * j2 + 1, n3 = 2 * j3 + 1;

    double C[7][7][7];
    for (int a = 0; a < 7; ++a) for (int b = 0; b < 7; ++b) for (int c = 0; c < 7; ++c) C[a][b][c] = 0.0;
    for (int m1 = -j1; m1 <= j1; ++m1)
        for (int m2 = -j2; m2 <= j2; ++m2) {
            int m3 = m1 + m2;
            if (m3 < -j3 || m3 > j3) continue;
            double pref = (double)(2 * j3 + 1) * dfact(j3 + j1 - j2) * dfact(j3 - j1 + j2) *
                          dfact(j1 + j2 - j3) / dfact(j1 + j2 + j3 + 1);
            pref *= dfact(j3 + m3) * dfact(j3 - m3) * dfact(j1 - m1) * dfact(j1 + m1) *
                    dfact(j2 - m2) * dfact(j2 + m2);
            int kmin = 0;
            if (-(j3 - j2 + m1) > kmin) kmin = -(j3 - j2 + m1);
            if (-(j3 - j1 - m2) > kmin) kmin = -(j3 - j1 - m2);
            int kmax = j1 + j2 - j3;
            if (j1 - m1 < kmax) kmax = j1 - m1;
            if (j2 + m2 < kmax) kmax = j2 + m2;
            double s = 0.0;
            for (int t = kmin; t <= kmax; ++t) {
                double d = dfact(t) * dfact(j1 + j2 - j3 - t) * dfact(j1 - m1 - t) *
                           dfact(j2 + m2 - t) * dfact(j3 - j2 + m1 + t) * dfact(j3 - j1 - m2 + t);
                s += ((t & 1) ? -1.0 : 1.0) / d;
            }
            C[m1 + j1][m2 + j2][m3 + j3] = sqrt(pref) * s;
        }

    cd Q1[7][7], Q2[7][7], Q3[7][7];
    qmat(j1, Q1); qmat(j2, Q2); qmat(j3, Q3);

    double Dr[7][7][7];
    bool ok = false;
    for (int pass = 0; pass < 2 && !ok; ++pass) {
        double maxim = 0.0;
        for (int i = 0; i < n1; ++i)
            for (int j = 0; j < n2; ++j)
                for (int kk = 0; kk < n3; ++kk) {
                    cd s; s.re = 0.0; s.im = 0.0;
                    for (int a = 0; a < n1; ++a)
                        for (int b = 0; b < n2; ++b) {
                            cd t1 = cmul(Q1[a][i], Q2[b][j]);
                            for (int c = 0; c < n3; ++c) {
                                double cc = C[a][b][c];
                                if (cc == 0.0) continue;
                                cd q = Q3[c][kk];
                                if (pass == 0) q.im = -q.im;   // conj(Q3) first
                                cd t2 = cmul(t1, q);
                                s.re += cc * t2.re; s.im += cc * t2.im;
                            }
                        }
                    Dr[i][j][kk] = s.re;
                    double ai = fabs(s.im); if (ai > maxim) maxim = ai;
                }
        if (maxim < 1e-9) ok = true;
    }
    double nrm = 0.0;
    for (int i = 0; i < n1; ++i) for (int j = 0; j < n2; ++j) for (int kk = 0; kk < n3; ++kk)
        nrm += Dr[i][j][kk] * Dr[i][j][kk];
    nrm = sqrt(nrm);
    float* outp = w3j_out + W3OFF[k];
    for (int i = 0; i < n1; ++i) for (int j = 0; j < n2; ++j) for (int kk = 0; kk < n3; ++kk)
        outp[(i * n2 + j) * n3 + kk] = (float)(Dr[i][j][kk] / nrm);
}

// ---------------------------------------------------------------------------
// Zero the scatter accumulators (128-bit stores)
// ---------------------------------------------------------------------------
__global__ void zero_kernel(float4* __restrict__ p, size_t n4) {
    size_t i = (size_t)blockIdx.x * blockDim.x + threadIdx.x;
    size_t stride = (size_t)gridDim.x * blockDim.x;
    const float4 z = make_float4(0.f, 0.f, 0.f, 0.f);
    for (; i < n4; i += stride) p[i] = z;
}

// ---------------------------------------------------------------------------
// Node linear: H[node][v*4+i] for i=0 (l=0, W0) and i=1..3 (l=1, W1)
// one wave per 16-node x 16-channel tile, K=128 via v_wmma_f32_16x16x4_f32
// ---------------------------------------------------------------------------
__global__ __launch_bounds__(32) void node_linear_kernel(
    const float* __restrict__ nf, const float* __restrict__ W0,
    const float* __restrict__ W1, float* __restrict__ H)
{
    const int lane  = threadIdx.x & 31;
    const int node0 = blockIdx.x * 16;
    const int v0    = blockIdx.y * 16;
    const int i     = blockIdx.z;            // 0..3
    const int kb    = (lane < 16) ? 0 : 2;   // A/B K sub-offset
    const int mk    = lane & 15;             // M for A rows / N for B cols
    const float scale = 0.08838834764831845f; // 1/sqrt(128)

    v8f acc = {0.f,0.f,0.f,0.f,0.f,0.f,0.f,0.f};
    const int node = node0 + mk;
    if (i == 0) {
        const float* __restrict__ abase = nf + (size_t)node * 512;
        for (int k0 = 0; k0 < 128; k0 += 4) {
            const int u = k0 + kb;
            v2f a = *(const v2f*)(abase + u);      // contiguous, 8B aligned
            v2f b;
            b[0] = W0[(size_t)(u + 0) * 128 + v0 + mk] * scale;
            b[1] = W0[(size_t)(u + 1) * 128 + v0 + mk] * scale;
            acc = wmma4(a, b, acc);
        }
    } else {
        const float* __restrict__ abase = nf + (size_t)node * 512 + 128 + (i - 1);
        for (int k0 = 0; k0 < 128; k0 += 4) {
            const int u = k0 + kb;
            v2f a, b;
            a[0] = abase[(u + 0) * 3];
            a[1] = abase[(u + 1) * 3];
            b[0] = W1[(size_t)(u + 0) * 128 + v0 + mk] * scale;
            b[1] = W1[(size_t)(u + 1) * 128 + v0 + mk] * scale;
            acc = wmma4(a, b, acc);
        }
    }
    const int mbase = (lane < 16) ? 0 : 8;
    #pragma unroll
    for (int r = 0; r < 8; ++r)
        H[(size_t)(node0 + mbase + r) * 512 + (size_t)(v0 + mk) * 4 + i] = acc[r];
}

// ---------------------------------------------------------------------------
// Fused edge kernel: radial MLP (WMMA, LDS-resident) + tensor product +
// atomic scatter to receiver accumulators. 256 threads / 16 edges per block.
// ---------------------------------------------------------------------------
__global__ __launch_bounds__(256) void edge_kernel(
    const float* __restrict__ edge_attrs, const float* __restrict__ edge_feats,
    const int* __restrict__ edge_index,
    const float* __restrict__ RW0, const float* __restrict__ RW1,
    const float* __restrict__ RW2, const float* __restrict__ RW3,
    const float* __restrict__ H, const float* __restrict__ w3j,
    float* __restrict__ A0, float* __restrict__ A1,
    float* __restrict__ A2, float* __restrict__ A3)
{
    extern __shared__ float smem[];
    float* s_feat = smem;              // 16*8    = 128
    float* s_sh   = s_feat + 128;      // 16*16   = 256
    float* s_w3j  = s_sh + 256;        // 402 (pad 416)
    float* s_c    = s_w3j + 416;       // 16*88   = 1408
    float* s_b0   = s_c + 1408;        // 16*64   = 1024
    float* s_b1   = s_b0 + 1024;       // 16*64   = 1024
    float* s_h    = s_b1 + 1024;       // 16*512  = 8192  (16B-aligned offset)
    float* s_r    = s_h + 8192;        // 16*1280 = 20480
    __shared__ int s_recv[16];

    const int tid = threadIdx.x;
    const int e0  = blockIdx.x * 16;

    // ---- gather tile inputs into LDS ----
    if (tid < 128) s_feat[tid] = edge_feats[(size_t)e0 * 8 + tid];
    s_sh[tid] = edge_attrs[(size_t)e0 * 16 + tid];
    if (tid < 16) s_recv[tid] = edge_index[NEDGES + e0 + tid];
    for (int idx = tid; idx < 402; idx += 256) s_w3j[idx] = w3j[idx];
    {   // sender features: 16 edges * 512 floats as 128-bit transfers
        const float4* __restrict__ Hv = (const float4*)H;
        float4* s_h4 = (float4*)s_h;
        #pragma unroll
        for (int it = 0; it < 8; ++it) {
            int idx = tid + it * 256;         // 16 edges * 128 float4
            int e = idx >> 7;
            int f = idx & 127;
            int snd = edge_index[e0 + e];
            s_h4[e * 128 + f] = Hv[(size_t)snd * 128 + f];
        }
    }
    __syncthreads();

    // ---- per-edge c[i][kk] = sqrt(2l3+1) * sum_j w3j[i][j][kk] * sh[j] ----
    {
        const int e = tid >> 4;
        const int w = tid & 15;
        const float* sh = s_sh + e * 16;
        float* ce = s_c + e * 88;
        const int L1T[10] = {0,1,0,1,1,0,1,1,0,1};
        const int L2T[10] = {0,1,1,0,2,2,1,3,3,2};
        const int L3T[10] = {0,0,1,1,1,2,2,2,3,3};
        const int W3OFF[10] = {0,1,10,19,28,73,98,143,248,297};
        const int COFF[10]  = {0,1,4,7,16,25,30,45,60,67};
        const int SHOFF[4]  = {0,1,4,9};
        const float SQ3[4]  = {1.0f, 1.7320508075688772f, 2.2360679774997896f, 2.6457513110645906f};
        #pragma unroll
        for (int k = 0; k < 10; ++k) {
            const int n1 = 2 * L1T[k] + 1, n2 = 2 * L2T[k] + 1, n3 = 2 * L3T[k] + 1;
            const float* w3  = s_w3j + W3OFF[k];
            const float* shk = sh + SHOFF[L2T[k]];
            const float sq = SQ3[L3T[k]];
            for (int idx = w; idx < n1 * n3; idx += 16) {
                int i = idx / n3, kk = idx % n3;
                float s = 0.f;
                for (int j = 0; j < n2; ++j) s += w3[(i * n2 + j) * n3 + kk] * shk[j];
                ce[COFF[k] + idx] = s * sq;
            }
        }
    }

    // ---- radial MLP via f32 WMMA; M = 16 edges ----
    const int wv    = tid >> 5;
    const int lane  = tid & 31;
    const int kb    = (lane < 16) ? 0 : 2;
    const int mk    = lane & 15;
    const int mbase = (lane < 16) ? 0 : 8;

    // L0: 8 -> 64 (silu)
    if (wv < 4) {
        const int c0 = wv * 16;
        const float sc = 0.35355339059327373f; // 1/sqrt(8)
        v8f acc = {0.f,0.f,0.f,0.f,0.f,0.f,0.f,0.f};
        for (int k0 = 0; k0 < 8; k0 += 4) {
            const int u = k0 + kb;
            v2f a = *(const v2f*)(s_feat + mk * 8 + u);
            v2f b;
            b[0] = RW0[(u + 0) * 64 + c0 + mk] * sc;
            b[1] = RW0[(u + 1) * 64 + c0 + mk] * sc;
            acc = wmma4(a, b, acc);
        }
        #pragma unroll
        for (int r = 0; r < 8; ++r) {
            float x = acc[r];
            s_b0[(mbase + r) * 64 + c0 + mk] = x / (1.f + __expf(-x));
        }
    }
    __syncthreads();

    // L1: 64 -> 64 (silu)
    if (wv < 4) {
        const int c0 = wv * 16;
        v8f acc = {0.f,0.f,0.f,0.f,0.f,0.f,0.f,0.f};
        for (int k0 = 0; k0 < 64; k0 += 4) {
            const int u = k0 + kb;
            v2f a = *(const v2f*)(s_b0 + mk * 64 + u);
            v2f b;
            b[0] = RW1[(u + 0) * 64 + c0 + mk] * 0.125f;
            b[1] = RW1[(u + 1) * 64 + c0 + mk] * 0.125f;
            acc = wmma4(a, b, acc);
        }
        #pragma unroll
        for (int r = 0; r < 8; ++r) {
            float x = acc[r];
            s_b1[(mbase + r) * 64 + c0 + mk] = x / (1.f + __expf(-x));
        }
    }
    __syncthreads();

    // L2: 64 -> 64 (silu)
    if (wv < 4) {
        const int c0 = wv * 16;
        v8f acc = {0.f,0.f,0.f,0.f,0.f,0.f,0.f,0.f};
        for (int k0 = 0; k0 < 64; k0 += 4) {
            const int u = k0 + kb;
            v2f a = *(const v2f*)(s_b1 + mk * 64 + u);
            v2f b;
            b[0] = RW2[(u + 0) * 64 + c0 + mk] * 0.125f;
            b[1] = RW2[(u + 1) * 64 + c0 + mk] * 0.125f;
            acc = wmma4(a, b, acc);
        }
        #pragma unroll
        for (int r = 0; r < 8; ++r) {
            float x = acc[r];
            s_b0[(mbase + r) * 64 + c0 + mk] = x / (1.f + __expf(-x));
        }
    }
    __syncthreads();

    // L3: 64 -> 1280 (no activation); 10 column tiles per wave
    for (int t = 0; t < 10; ++t) {
        const int c0 = (wv + 8 * t) * 16;
        v8f acc = {0.f,0.f,0.f,0.f,0.f,0.f,0.f,0.f};
        for (int k0 = 0; k0 < 64; k0 += 4) {
            const int u = k0 + kb;
            v2f a = *(const v2f*)(s_b0 + mk * 64 + u);
            v2f b;
            b[0] = RW3[(size_t)(u + 0) * 1280 + c0 + mk] * 0.125f;
            b[1] = RW3[(size_t)(u + 1) * 1280 + c0 + mk] * 0.125f;
            acc = wmma4(a, b, acc);
        }
        #pragma unroll
        for (int r = 0; r < 8; ++r)
            s_r[(mbase + r) * 1280 + c0 + mk] = acc[r];
    }
    __syncthreads();

    // ---- tensor product + atomic scatter: thread handles (edge,u) pairs ----
    {
        const int L1T[10] = {0,1,0,1,1,0,1,1,0,1};
        const int L3T[10] = {0,0,1,1,1,2,2,2,3,3};
        const int COFF[10] = {0,1,4,7,16,25,30,45,60,67};
        const int COLBASE[10] = {0,128,0,128,256,0,128,256,0,128};
        const int COLS[4] = {256,384,384,256};
        float* const ACC[4] = {A0, A1, A2, A3};
        #pragma unroll
        for (int it = 0; it < 8; ++it) {
            const int p = tid + it * 256;     // 16*128 pairs
            const int e = p >> 7;
            const int u = p & 127;
            const float4 hv = *(const float4*)(s_h + e * 512 + u * 4); // ds_load_b128
            const float h0  = hv.x;
            const float h1a = hv.y, h1b = hv.z, h1c = hv.w;
            const float* ce = s_c + e * 88;
            const float* re = s_r + e * 1280;
            const int recv = s_recv[e];
            #pragma unroll
            for (int k = 0; k < 10; ++k) {
                const int l1 = L1T[k], l3 = L3T[k], n3 = 2 * l3 + 1;
                const float wgt = re[k * 128 + u];
                const float* c = ce + COFF[k];
                float* dst = ACC[l3] +
                    ((size_t)recv * COLS[l3] + COLBASE[k] + u) * n3;
                #pragma unroll
                for (int kk = 0; kk < n3; ++kk) {
                    float s;
                    if (l1 == 0) s = c[kk] * h0;
                    else s = c[kk] * h1a + c[n3 + kk] * h1b + c[2 * n3 + kk] * h1c;
                    __hip_atomic_fetch_add(dst + kk, s * wgt,
                                           __ATOMIC_RELAXED, __HIP_MEMORY_SCOPE_AGENT);
                }
            }
        }
    }
}

// ---------------------------------------------------------------------------
// Output linears: per (l3,kk) slot, GEMM (N x cols) @ (cols x 128) via WMMA
// ---------------------------------------------------------------------------
__global__ __launch_bounds__(32) void out_linear_kernel(
    const float* __restrict__ A0, const float* __restrict__ A1,
    const float* __restrict__ A2, const float* __restrict__ A3,
    const float* __restrict__ Wo0, const float* __restrict__ Wo1,
    const float* __restrict__ Wo2, const float* __restrict__ Wo3,
    float* __restrict__ out)
{
    const int SLOT_L3[16] = {0,1,1,1,2,2,2,2,2,3,3,3,3,3,3,3};
    const int SLOT_KK[16] = {0,0,1,2,0,1,2,3,4,0,1,2,3,4,5,6};
    const int COLS[4] = {256,384,384,256};
    const float SCLT[4] = {0.00390625f, 0.0031894443f, 0.0031894443f, 0.00390625f};

    const int lane  = threadIdx.x & 31;
    const int node0 = blockIdx.x * 16;
    const int v0    = blockIdx.y * 16;
    const int slot  = blockIdx.z;
    const int l3    = SLOT_L3[slot];
    const int kk    = SLOT_KK[slot];
    const int n3    = 2 * l3 + 1;
    const int cols  = COLS[l3];
    const float scl = SCLT[l3];
    const float* Acc = (l3 == 0) ? A0 : (l3 == 1) ? A1 : (l3 == 2) ? A2 : A3;
    const float* Wo  = (l3 == 0) ? Wo0 : (l3 == 1) ? Wo1 : (l3 == 2) ? Wo2 : Wo3;

    const int kb = (lane < 16) ? 0 : 2;
    const int mk = lane & 15;

    v8f acc = {0.f,0.f,0.f,0.f,0.f,0.f,0.f,0.f};
    for (int k0 = 0; k0 < cols; k0 += 4) {
        const int col = k0 + kb;
        v2f a, b;
        a[0] = Acc[((size_t)(node0 + mk) * cols + col + 0) * n3 + kk];
        a[1] = Acc[((size_t)(node0 + mk) * cols + col + 1) * n3 + kk];
        b[0] = Wo[(size_t)(col + 0) * 128 + v0 + mk] * scl;
        b[1] = Wo[(size_t)(col + 1) * 128 + v0 + mk] * scl;
        acc = wmma4(a, b, acc);
    }
    const int mbase = (lane < 16) ? 0 : 8;
    #pragma unroll
    for (int r = 0; r < 8; ++r)
        out[((size_t)(node0 + mbase + r) * 128 + v0 + mk) * 16 + slot] = acc[r];
}

// ---------------------------------------------------------------------------
extern "C" void kernel_launch(void* const* d_in, const int* in_sizes, int n_in,
                              void* d_out, int out_size, void* d_ws, size_t ws_size,
                              hipStream_t stream) {
    (void)in_sizes; (void)n_in; (void)out_size; (void)ws_size;
    const float* node_feats = (const float*)d_in[0];
    // d_in[1] = node_attrs (unused by reference)
    const float* edge_attrs = (const float*)d_in[2];
    const float* edge_feats = (const float*)d_in[3];
    const int*   edge_index = (const int*)d_in[4];
    const float* W0  = (const float*)d_in[5];
    const float* W1  = (const float*)d_in[6];
    const float* RW0 = (const float*)d_in[7];
    const float* RW1 = (const float*)d_in[8];
    const float* RW2 = (const float*)d_in[9];
    const float* RW3 = (const float*)d_in[10];
    const float* Wo0 = (const float*)d_in[11];
    const float* Wo1 = (const float*)d_in[12];
    const float* Wo2 = (const float*)d_in[13];
    const float* Wo3 = (const float*)d_in[14];

    // workspace layout (floats)
    float* ws  = (float*)d_ws;
    float* w3j = ws;                        // 402 (pad to 512)
    float* H   = ws + 512;                  // 10000*512       = 5,120,000
    float* A0  = H  + 5120000;              // 10000*256*1     = 2,560,000
    float* A1  = A0 + 2560000;              // 10000*384*3     = 11,520,000
    float* A2  = A1 + 11520000;             // 10000*384*5     = 19,200,000
    float* A3  = A2 + 19200000;             // 10000*256*7     = 17,920,000

    w3j_kernel<<<1, 32, 0, stream>>>(w3j);

    const size_t accN = 2560000ull + 11520000ull + 19200000ull + 17920000ull;
    zero_kernel<<<2048, 256, 0, stream>>>((float4*)A0, accN / 4);

    node_linear_kernel<<<dim3(625, 8, 4), 32, 0, stream>>>(node_feats, W0, W1, H);

    const size_t shbytes = 32928 * sizeof(float); // ~128.6 KB LDS / block
    edge_kernel<<<dim3(NEDGES / 16), 256, shbytes, stream>>>(
        edge_attrs, edge_feats, edge_index, RW0, RW1, RW2, RW3, H, w3j,
        A0, A1, A2, A3);

    out_linear_kernel<<<dim3(625, 8, 16), 32, 0, stream>>>(
        A0, A1, A2, A3, Wo0, Wo1, Wo2, Wo3, (float*)d_out);
}